// MHA_37915971289586
// MI455X (gfx1250) — compile-verified
//
#include <hip/hip_runtime.h>

typedef __attribute__((ext_vector_type(2))) float v2f;
typedef __attribute__((ext_vector_type(8))) float v8f;
typedef __attribute__((ext_vector_type(4))) unsigned int u32x4;
typedef __attribute__((ext_vector_type(4))) int i32x4;
typedef __attribute__((ext_vector_type(8))) int i32x8;

#define NUM_HEADS 16
#define HEAD_DIM  64
#define D_MODEL   1024
#define D_QKV     3072
#define SEQ_T     2048
#define NBATCH    2

// Tensor Data Mover availability (device pass only; host pass falls back)
#if __has_builtin(__builtin_amdgcn_tensor_load_to_lds) && __has_builtin(__builtin_amdgcn_s_wait_tensorcnt)
#define USE_TDM 1
#else
#define USE_TDM 0
#endif

static __device__ __forceinline__ v8f wmma_f32(v2f a, v2f b, v8f c) {
    // D = A(16x4) * B(4x16) + C(16x16), all f32 (CDNA5 V_WMMA_F32_16X16X4_F32)
    return __builtin_amdgcn_wmma_f32_16x16x4_f32(
        /*neg_a=*/false, a, /*neg_b=*/false, b,
        /*c_mod=*/(short)0, c, /*reuse_a=*/false, /*reuse_b=*/false);
}

#if USE_TDM
// Issue one TDM 2D-tile DMA: global (row-major, row pitch = stride0 elements,
// fp32) -> LDS, tile_d1 rows x tile_d0 elements, packed contiguously in LDS.
// D# bit layout per CDNA5 ISA ch.8 (group0: count/lds/global/type; group1:
// data_size, tensor dims, tile dims, dim0 stride). Groups 2/3 zero (2D).
static __device__ __forceinline__ void tdm_load_2d(
    unsigned lds_byte_addr, const void* gaddr,
    unsigned tensor_d0, unsigned tensor_d1,
    unsigned tile_d0, unsigned tile_d1, unsigned stride0)
{
    unsigned long long ga = (unsigned long long)(uintptr_t)gaddr;
    u32x4 g0;
    g0[0] = 1u;                                        // count=1 (valid user D#)
    g0[1] = lds_byte_addr;                             // lds_addr [63:32]
    g0[2] = (unsigned)(ga & 0xFFFFFFFFu);              // global_addr lo
    g0[3] = (unsigned)((ga >> 32) & 0x01FFFFFFu)       // global_addr hi (57b)
            | 0x80000000u;                             // type=2 ("image")
    i32x8 g1;
    g1[0] = (int)(2u << 16);                           // data_size=2 -> 4 bytes
    g1[1] = (int)(tensor_d0 << 16);                    // tensor_dim0[15:0]
    g1[2] = (int)((tensor_d0 >> 16) | (tensor_d1 << 16));
    g1[3] = (int)((tensor_d1 >> 16) | (tile_d0 << 16)); // tile_dim0
    g1[4] = (int)(tile_d1 & 0xFFFFu);                  // tile_dim1 (tile_dim2=0)
    g1[5] = (int)stride0;                              // tensor_dim0_stride lo
    g1[6] = 0;
    g1[7] = 0;
    i32x4 z4 = {0, 0, 0, 0};
#if __clang_major__ >= 23
    i32x8 z8 = {0, 0, 0, 0, 0, 0, 0, 0};
    __builtin_amdgcn_tensor_load_to_lds(g0, g1, z4, z4, z8, 0);
#else
    __builtin_amdgcn_tensor_load_to_lds(g0, g1, z4, z4, 0);
#endif
}
#endif

// ---------------------------------------------------------------------------
// Stage 1: QKV projection  C[4096,3072] = X[4096,1024] @ W[1024,3072] + b
// WG = 256 threads (8 waves). WG tile 64(M) x 128(N). Wave tile 32x32.
// X/W tiles double-buffered in LDS via TDM (fallback: sync copies).
// ---------------------------------------------------------------------------
__global__ __launch_bounds__(256, 2)
void qkv_gemm_kernel(const float* __restrict__ X, const float* __restrict__ W,
                     const float* __restrict__ bias,
                     float* __restrict__ Qo, float* __restrict__ Ko,
                     float* __restrict__ Vo)
{
    __shared__ float Xs[2][64 * 16];    // 64 rows x 16 K
    __shared__ float Ws[2][16 * 128];   // 16 K x 128 cols

    const int tid  = threadIdx.x;
    const int wave = tid >> 5;
    const int lane = tid & 31;
    const int lh   = lane >> 4;
    const int ln   = lane & 15;

    const int wg_m = blockIdx.x * 64;
    const int wg_n = blockIdx.y * 128;
    const int wm   = (wave >> 2) * 32;
    const int wn   = (wave & 3) * 32;

    auto load_tiles = [&](int k0, int buf) {
#if USE_TDM
        if (wave == 0) {
            tdm_load_2d((unsigned)(uintptr_t)&Xs[buf][0],
                        X + (size_t)wg_m * D_MODEL + k0,
                        /*tensor_d0=*/D_MODEL, /*tensor_d1=*/4096,
                        /*tile_d0=*/16, /*tile_d1=*/64, /*stride0=*/D_MODEL);
            tdm_load_2d((unsigned)(uintptr_t)&Ws[buf][0],
                        W + (size_t)k0 * D_QKV + wg_n,
                        /*tensor_d0=*/D_QKV, /*tensor_d1=*/D_MODEL,
                        /*tile_d0=*/128, /*tile_d1=*/16, /*stride0=*/D_QKV);
        }
#else
        {
            const int r  = tid >> 2;
            const int c4 = tid & 3;
            ((float4*)Xs[buf])[tid] =
                ((const float4*)(X + (size_t)(wg_m + r) * D_MODEL + k0))[c4];
        }
#pragma unroll
        for (int i = 0; i < 2; ++i) {
            const int idx = tid + i * 256;
            const int r   = idx >> 5;
            const int c4  = idx & 31;
            ((float4*)Ws[buf])[idx] =
                ((const float4*)(W + (size_t)(k0 + r) * D_QKV + wg_n))[c4];
        }
#endif
    };
    auto wait_tiles = [&]() {
#if USE_TDM
        if (wave == 0) __builtin_amdgcn_s_wait_tensorcnt(0);
#endif
    };

    v8f acc[2][2] = {};

    load_tiles(0, 0);
    wait_tiles();
    __syncthreads();

    int cur = 0;
    for (int k0 = 0; k0 < D_MODEL; k0 += 16) {
        const int nxt = cur ^ 1;
        if (k0 + 16 < D_MODEL) load_tiles(k0 + 16, nxt);

        const float* Xc = Xs[cur];
        const float* Wc = Ws[cur];
#pragma unroll
        for (int kk = 0; kk < 16; kk += 4) {
            v2f a0 = *(const v2f*)&Xc[(wm + ln) * 16 + kk + 2 * lh];
            v2f a1 = *(const v2f*)&Xc[(wm + 16 + ln) * 16 + kk + 2 * lh];
            v2f b0, b1;
            b0[0] = Wc[(kk + 2 * lh + 0) * 128 + wn + ln];
            b0[1] = Wc[(kk + 2 * lh + 1) * 128 + wn + ln];
            b1[0] = Wc[(kk + 2 * lh + 0) * 128 + wn + 16 + ln];
            b1[1] = Wc[(kk + 2 * lh + 1) * 128 + wn + 16 + ln];

            acc[0][0] = wmma_f32(a0, b0, acc[0][0]);
            acc[0][1] = wmma_f32(a0, b1, acc[0][1]);
            acc[1][0] = wmma_f32(a1, b0, acc[1][0]);
            acc[1][1] = wmma_f32(a1, b1, acc[1][1]);
        }
        wait_tiles();
        __syncthreads();
        cur = nxt;
    }

    // Epilogue: bias + scatter into Q/K/V [N,H,T,D]; Q scaled by 1/sqrt(64).
#pragma unroll
    for (int i = 0; i < 2; ++i)
#pragma unroll
        for (int j = 0; j < 2; ++j) {
            const int m0 = wg_m + wm + i * 16;
            const int n0 = wg_n + wn + j * 16;
#pragma unroll
            for (int v = 0; v < 8; ++v) {
                const int m = m0 + v + 8 * lh;
                const int c = n0 + ln;
                float val   = acc[i][j][v] + bias[c];
                const int s   = c % 3;
                const int rem = c / 3;
                const int d   = rem & 63;
                const int h   = rem >> 6;
                const int nb  = m >> 11;
                const int t   = m & 2047;
                const size_t off =
                    (((size_t)nb * NUM_HEADS + h) * SEQ_T + t) * HEAD_DIM + d;
                if (s == 0)      Qo[off] = val * 0.125f;
                else if (s == 1) Ko[off] = val;
                else             Vo[off] = val;
            }
        }
}

// ---------------------------------------------------------------------------
// Stage 2: flash attention per (n,h). WG = 128 threads (4 waves), 64 query
// rows per WG. K/V blocks of 64 double-buffered in LDS via TDM.
// ---------------------------------------------------------------------------
__global__ __launch_bounds__(128, 2)
void attn_kernel(const float* __restrict__ Q, const float* __restrict__ K,
                 const float* __restrict__ V, float* __restrict__ out)
{
    __shared__ float Qs[4][16 * 64];
    __shared__ float Ks[2][64 * 64];
    __shared__ float Vs[2][64 * 64];
    __shared__ float Ps[4][16 * 64];

    const int tid  = threadIdx.x;
    const int wave = tid >> 5;
    const int lane = tid & 31;
    const int lh   = lane >> 4;
    const int ln   = lane & 15;

    const int q0 = blockIdx.x * 64;
    const int h  = blockIdx.y;
    const int nb = blockIdx.z;

    const size_t bh = ((size_t)nb * NUM_HEADS + h) * SEQ_T * HEAD_DIM;
    const float* Qb = Q + bh;
    const float* Kb = K + bh;
    const float* Vb = V + bh;

    auto load_kv = [&](int kb0, int buf) {
#if USE_TDM
        if (wave == 0) {
            tdm_load_2d((unsigned)(uintptr_t)&Ks[buf][0],
                        Kb + (size_t)kb0 * HEAD_DIM,
                        /*tensor_d0=*/HEAD_DIM, /*tensor_d1=*/SEQ_T,
                        /*tile_d0=*/64, /*tile_d1=*/64, /*stride0=*/HEAD_DIM);
            tdm_load_2d((unsigned)(uintptr_t)&Vs[buf][0],
                        Vb + (size_t)kb0 * HEAD_DIM,
                        /*tensor_d0=*/HEAD_DIM, /*tensor_d1=*/SEQ_T,
                        /*tile_d0=*/64, /*tile_d1=*/64, /*stride0=*/HEAD_DIM);
        }
#else
        const float4* ksrc = (const float4*)(Kb + (size_t)kb0 * HEAD_DIM);
        const float4* vsrc = (const float4*)(Vb + (size_t)kb0 * HEAD_DIM);
        for (int i = tid; i < 1024; i += 128) {
            ((float4*)Ks[buf])[i] = ksrc[i];
            ((float4*)Vs[buf])[i] = vsrc[i];
        }
#endif
    };
    auto wait_kv = [&]() {
#if USE_TDM
        if (wave == 0) __builtin_amdgcn_s_wait_tensorcnt(0);
#endif
    };

    // Per-wave 16x64 Q tile (same-wave DS ordering; no barrier needed)
    {
        const float4* src =
            (const float4*)(Qb + (size_t)(q0 + wave * 16) * HEAD_DIM);
        for (int i = lane; i < 256; i += 32)
            ((float4*)Qs[wave])[i] = src[i];
    }

    v8f o[4] = {};
    float mrow[8], lrow[8];
#pragma unroll
    for (int v = 0; v < 8; ++v) { mrow[v] = -1.0e30f; lrow[v] = 0.0f; }

    load_kv(0, 0);
    wait_kv();
    __syncthreads();

    int cur = 0;
    for (int kb0 = 0; kb0 < SEQ_T; kb0 += 64) {
        const int nxt = cur ^ 1;
        if (kb0 + 64 < SEQ_T) load_kv(kb0 + 64, nxt);

        const float* Kc = Ks[cur];
        const float* Vc = Vs[cur];

        // S = Q * K^T  (16x64 per wave, 4 column frags)
        v8f s[4] = {};
#pragma unroll
        for (int d0 = 0; d0 < HEAD_DIM; d0 += 4) {
            v2f a = *(const v2f*)&Qs[wave][ln * 64 + d0 + 2 * lh];
#pragma unroll
            for (int f = 0; f < 4; ++f) {
                v2f b = *(const v2f*)&Kc[(f * 16 + ln) * 64 + d0 + 2 * lh];
                s[f] = wmma_f32(a, b, s[f]);
            }
        }

        // Online softmax (C-frag row v + 8*lh lives within one 16-lane half)
        float scale[8];
#pragma unroll
        for (int v = 0; v < 8; ++v) {
            float mx = fmaxf(fmaxf(s[0][v], s[1][v]), fmaxf(s[2][v], s[3][v]));
            mx = fmaxf(mx, __shfl_xor(mx, 1, 32));
            mx = fmaxf(mx, __shfl_xor(mx, 2, 32));
            mx = fmaxf(mx, __shfl_xor(mx, 4, 32));
            mx = fmaxf(mx, __shfl_xor(mx, 8, 32));
            const float mnew = fmaxf(mrow[v], mx);
            scale[v] = __expf(mrow[v] - mnew);
            mrow[v]  = mnew;
        }
#pragma unroll
        for (int v = 0; v < 8; ++v) {
            float rs = 0.0f;
#pragma unroll
            for (int f = 0; f < 4; ++f) {
                const float p = __expf(s[f][v] - mrow[v]);
                s[f][v] = p;
                rs += p;
            }
            rs += __shfl_xor(rs, 1, 32);
            rs += __shfl_xor(rs, 2, 32);
            rs += __shfl_xor(rs, 4, 32);
            rs += __shfl_xor(rs, 8, 32);
            lrow[v] = lrow[v] * scale[v] + rs;
#pragma unroll
            for (int f = 0; f < 4; ++f) o[f][v] *= scale[v];
        }

        // Spill P (C-layout) to row-major LDS, re-read as A frags (same wave)
#pragma unroll
        for (int f = 0; f < 4; ++f)
#pragma unroll
            for (int v = 0; v < 8; ++v)
                Ps[wave][(v + 8 * lh) * 64 + f * 16 + ln] = s[f][v];

        // O += P * V
#pragma unroll
        for (int k = 0; k < 64; k += 4) {
            v2f a = *(const v2f*)&Ps[wave][ln * 64 + k + 2 * lh];
#pragma unroll
            for (int f = 0; f < 4; ++f) {
                v2f b;
                b[0] = Vc[(k + 2 * lh + 0) * 64 + f * 16 + ln];
                b[1] = Vc[(k + 2 * lh + 1) * 64 + f * 16 + ln];
                o[f] = wmma_f32(a, b, o[f]);
            }
        }

        wait_kv();
        __syncthreads();
        cur = nxt;
    }

    // Normalize and store: out[(n*T + t)*1024 + h*64 + d]
    float inv[8];
#pragma unroll
    for (int v = 0; v < 8; ++v) inv[v] = 1.0f / lrow[v];
#pragma unroll
    for (int f = 0; f < 4; ++f)
#pragma unroll
        for (int v = 0; v < 8; ++v) {
            const int t = q0 + wave * 16 + v + 8 * lh;
            const int d = f * 16 + ln;
            out[((size_t)nb * SEQ_T + t) * D_MODEL + h * HEAD_DIM + d] =
                o[f][v] * inv[v];
        }
}

// ---------------------------------------------------------------------------
extern "C" void kernel_launch(void* const* d_in, const int* in_sizes, int n_in,
                              void* d_out, int out_size, void* d_ws, size_t ws_size,
                              hipStream_t stream)
{
    const float* x    = (const float*)d_in[0];
    // d_in[1] = mask: adds a per-query-row constant over the key axis ->
    // softmax is shift-invariant per row, so it is a mathematical no-op.
    const float* Wqkv = (const float*)d_in[2];
    const float* bqkv = (const float*)d_in[3];
    float* out = (float*)d_out;

    const size_t per = (size_t)NBATCH * NUM_HEADS * SEQ_T * HEAD_DIM;
    float* Qw = (float*)d_ws;
    float* Kw = Qw + per;
    float* Vw = Kw + per;

    qkv_gemm_kernel<<<dim3(4096 / 64, D_QKV / 128), 256, 0, stream>>>(
        x, Wqkv, bqkv, Qw, Kw, Vw);

    attn_kernel<<<dim3(SEQ_T / 64, NUM_HEADS, NBATCH), 128, 0, stream>>>(
        Qw, Kw, Vw, out);
}